// SDCLinear_12103217840599
// MI455X (gfx1250) — compile-verified
//
#include <hip/hip_runtime.h>

typedef __attribute__((ext_vector_type(2))) float v2f;
typedef __attribute__((ext_vector_type(8))) float v8f;

#define T_    64
#define N_    16
#define C_    64
#define DOUT_ 256

// One block per (n,c). 256 threads = 8 waves (wave32).
// Wave w computes o in [32w, 32w+32): 2 column tiles x 4 row tiles of Y = L @ S,
// L[t,s] = 0.5^(t-s) (s<=t), S[s,o] = in0[(s-D0(o))&63] + in1[(s-D1(o))&63].
__global__ __launch_bounds__(256)
void sdc_wmma_kernel(const float* __restrict__ in,
                     const float* __restrict__ delay_raw,
                     const float* __restrict__ wscal,
                     float* __restrict__ out) {
  __shared__ float s_in[2][T_];
  __shared__ int   s_md[2];

  const int tid  = threadIdx.x;
  const int lane = tid & 31;
  const int wave = tid >> 5;
  const int nc   = blockIdx.x;
  const int n    = nc >> 6;   // / C_
  const int c    = nc & 63;   // % C_

  // Stage input[t,n,c,i] (T_*2 floats) into LDS.
  if (tid < 2 * T_) {
    const int t = tid >> 1, i = tid & 1;
    s_in[i][t] = in[t * (N_ * C_ * 2) + n * (C_ * 2) + c * 2 + i];
  }
  __syncthreads();

  // First-occurrence argmax over t per channel -> max allowed delay 63-argmax.
  if (tid < 2) {
    float mx = s_in[tid][0];
    int am = 0;
    for (int t = 1; t < T_; ++t) {
      float v = s_in[tid][t];
      if (v > mx) { mx = v; am = t; }
    }
    s_md[tid] = (T_ - 1) - am;
  }
  __syncthreads();

  const int   maxd0 = s_md[0];
  const int   maxd1 = s_md[1];
  const float w     = wscal[0];

  const int col   = lane & 15;          // N (=o) index within tile / M index for A
  const int sbase = (lane >> 4) << 1;   // K offset: 0 for lanes 0-15, 2 for 16-31
  const int rofs  = (lane >> 4) << 3;   // C/D row offset: +8 for hi half-wave

  // Per-lane clamped delays for the two o-tiles this wave owns.
  // _delay is integral for this input => stochastic rounding == floor.
  int d0[2], d1[2];
#pragma unroll
  for (int j = 0; j < 2; ++j) {
    const int o  = (2 * wave + j) * 16 + col;
    const int r0 = (int)floorf(fmaxf(delay_raw[o], 0.0f));
    const int r1 = (int)floorf(fmaxf(delay_raw[(DOUT_ - 1) - o], 0.0f));
    d0[j] = min(r0, maxd0);
    d1[j] = min(r1, maxd1);
  }

  v8f acc[4][2];
#pragma unroll
  for (int mt = 0; mt < 4; ++mt)
#pragma unroll
    for (int j = 0; j < 2; ++j)
      acc[mt][j] = (v8f)(0.0f);

  // K loop: 16 chunks of K=4 cover s = 0..63.
  for (int kc = 0; kc < 16; ++kc) {
    const int s0 = kc * 4 + sbase;
    const int s1 = s0 + 1;

    // B fragments (4x16, K x N): delayed gather + D_IN-sum fused.
    v2f b[2];
#pragma unroll
    for (int j = 0; j < 2; ++j) {
      b[j].x = s_in[0][(s0 - d0[j] + 64) & 63] + s_in[1][(s0 - d1[j] + 64) & 63];
      b[j].y = s_in[0][(s1 - d0[j] + 64) & 63] + s_in[1][(s1 - d1[j] + 64) & 63];
    }

#pragma unroll
    for (int mt = 0; mt < 4; ++mt) {
      // A fragment (16x4, M x K): L[t,s] = 2^(s-t) for s<=t else 0.
      const int trow = mt * 16 + col;
      v2f a;
      a.x = (s0 <= trow) ? ldexpf(1.0f, s0 - trow) : 0.0f;
      a.y = (s1 <= trow) ? ldexpf(1.0f, s1 - trow) : 0.0f;
#pragma unroll
      for (int j = 0; j < 2; ++j) {
        acc[mt][j] = __builtin_amdgcn_wmma_f32_16x16x4_f32(
            false, a, false, b[j], (short)0, acc[mt][j], false, false);
      }
    }
  }

  // Store: VGPR v holds rows (mt*16 + v) lanes 0-15 and (mt*16 + v + 8) lanes 16-31.
#pragma unroll
  for (int mt = 0; mt < 4; ++mt) {
#pragma unroll
    for (int j = 0; j < 2; ++j) {
      const int o = (2 * wave + j) * 16 + col;
#pragma unroll
      for (int v = 0; v < 8; ++v) {
        const int t = mt * 16 + v + rofs;
        out[(size_t)t * (N_ * C_ * DOUT_) + n * (C_ * DOUT_) + c * DOUT_ + o] =
            w * acc[mt][j][v];
      }
    }
  }
}

extern "C" void kernel_launch(void* const* d_in, const int* in_sizes, int n_in,
                              void* d_out, int out_size, void* d_ws, size_t ws_size,
                              hipStream_t stream) {
  const float* in        = (const float*)d_in[0];   // (T,N,C,D_IN) f32
  const float* delay_raw = (const float*)d_in[1];   // (D_OUT,1) f32
  const float* wscal     = (const float*)d_in[2];   // scalar f32
  float*       out       = (float*)d_out;           // (T,N,C,D_OUT) f32

  sdc_wmma_kernel<<<dim3(N_ * C_), dim3(256), 0, stream>>>(in, delay_raw, wscal, out);
}